// SemanticLoss_33526514713052
// MI455X (gfx1250) — compile-verified
//
#include <hip/hip_runtime.h>
#include <hip/hip_bf16.h>

typedef __attribute__((ext_vector_type(2))) float v2f;
typedef __attribute__((ext_vector_type(8))) float v8f;

#define NLAB 200
#define NHID 64
#define LPAD 65                 // 65-float pitch: label shifts bank by 1 -> conflict-light LDS atomics
#define SSEG 13000              // 200*64 sums + 200 counts per partial block
#define NCELLS_MAX 200000.0f
#define MIN_GROUP 5.0f

// ---------------- Phase 1: streaming segment partial sums (bandwidth bound) ----------------
__global__ __launch_bounds__(256) void seg_partial(const float* __restrict__ latent,
                                                   const int* __restrict__ labels,
                                                   float* __restrict__ partial, int nrows) {
  __shared__ float shs[NLAB * LPAD];   // 52,000 B
  __shared__ float shc[NLAB];
  const int tid = threadIdx.x;
  for (int i = tid; i < NLAB * LPAD; i += 256) shs[i] = 0.0f;
  for (int i = tid; i < NLAB; i += 256) shc[i] = 0.0f;
  __syncthreads();

  const int lane16 = tid & 15;
  const int rowoff = tid >> 4;         // 16 rows per block-iteration
  const int h4 = lane16 * 4;
  const long step = (long)gridDim.x * 16;

  for (long base = (long)blockIdx.x * 16; base < nrows; base += step) {
    long row = base + rowoff;
    // prefetch next tile while we work on this one
    long nrow = row + step;
    if (nrow < nrows)
      __builtin_prefetch(latent + nrow * NHID + h4, 0, 3);
    if (row < nrows) {
      const int lab = labels[row];
      const float4 v = *(const float4*)(latent + row * NHID + h4);
      float* p = &shs[lab * LPAD + h4];
      atomicAdd(p + 0, v.x);
      atomicAdd(p + 1, v.y);
      atomicAdd(p + 2, v.z);
      atomicAdd(p + 3, v.w);
      if (lane16 == 0) atomicAdd(&shc[lab], 1.0f);
    }
  }
  __syncthreads();

  float* out = partial + (long)blockIdx.x * SSEG;
  for (int i = tid; i < NLAB * NHID; i += 256)
    out[i] = shs[(i >> 6) * LPAD + (i & 63)];     // compact [label][hidden]
  for (int i = tid; i < NLAB; i += 256)
    out[NLAB * NHID + i] = shc[i];
}

// ---------------- Phase 2: reduce per-block partials (L2-resident) ----------------
__global__ __launch_bounds__(256) void reduce_partials(const float* __restrict__ partials,
                                                       float* __restrict__ reduced, int nblk) {
  const int e = blockIdx.x * blockDim.x + threadIdx.x;   // 0..25999
  if (e >= 2 * SSEG) return;
  const int half = (e >= SSEG) ? 1 : 0;
  const int idx = e - half * SSEG;
  const float* base = partials + (long)half * nblk * SSEG + idx;
  float s = 0.0f;
  for (int b = 0; b < nblk; ++b) s += base[(long)b * SSEG];
  reduced[e] = s;
}

// ---------------- Phase 3: centroid update + WMMA MSE + KL ----------------
__global__ __launch_bounds__(256) void finalize(const float* __restrict__ reduced,
                                                const float* __restrict__ cp0,
                                                const float* __restrict__ pc0,
                                                const float* __restrict__ ct0,
                                                const float* __restrict__ tc0,
                                                const float* __restrict__ ncells,
                                                float* __restrict__ out) {
  __shared__ float diffm[NHID * NLAB];  // 51,200 B, diff[h][l] row-major (64x200)
  __shared__ float s_np[NLAB], s_nt[NLAB], s_pc[NLAB], s_tc[NLAB], s_pcn[NLAB];
  __shared__ float s_diag[32];
  __shared__ float s_flag;
  const int tid = threadIdx.x;

  for (int l = tid; l < NLAB; l += 256) {
    s_np[l] = reduced[NLAB * NHID + l];
    s_nt[l] = reduced[SSEG + NLAB * NHID + l];
    s_pc[l] = pc0[l];
    s_tc[l] = tc0[l];
  }
  __syncthreads();

  if (tid == 0) {
    float m = -1.0f;
    for (int l = 0; l < NLAB; ++l) m = fmaxf(m, s_pc[l]);
    s_flag = (m >= NCELLS_MAX) ? 1.0f : 0.0f;
  }
  __syncthreads();
  const bool reset = s_flag > 0.5f;

  for (int l = tid; l < NLAB; l += 256) {
    const float pc = reset ? 1.0f : s_pc[l];
    const float np = s_np[l];
    s_pcn[l] = (np > MIN_GROUP) ? (pc + np) : pc;
  }

  for (int i = tid; i < NHID * NLAB; i += 256) {
    const int h = i / NLAB, l = i - h * NLAB;       // centroids are [64][200] row-major
    const float np = s_np[l], nt = s_nt[l];
    const float pc = reset ? 1.0f : s_pc[l];
    const float tc = s_tc[l];
    const float cp = cp0[i], ct = ct0[i];
    const float sp = reduced[l * NHID + h];
    const float st = reduced[SSEG + l * NHID + h];
    const float meanp = sp / fmaxf(np, 1.0f);
    const float meant = st / fmaxf(nt, 1.0f);
    const float cpn = (np > MIN_GROUP) ? (cp * pc + meanp * np) / (pc + np) : cp;
    const float ctn = (nt > MIN_GROUP) ? (ct * tc + meant * nt) / (tc + nt) : ct;
    diffm[i] = cpn - ctn;
  }
  __syncthreads();

  // Wave 0: MSE sum-of-squares via V_WMMA_F32_16X16X4_F32.
  // A (16x4 tile of diff rows) and B = A^T share the identical per-lane register
  // layout, so the same v2f feeds both operands. Diagonal of A*A^T accumulates
  // sum(diff^2) exactly in f32.
  if (tid < 32) {
    const int L = tid;
    const int m15 = L & 15;
    const int kh = (L >> 4) * 2;   // lanes 0-15: K=0,1 ; lanes 16-31: K=2,3
    v8f c = {0.f, 0.f, 0.f, 0.f, 0.f, 0.f, 0.f, 0.f};
    for (int t = 0; t < 4; ++t) {          // 4 tiles of 16 H-rows
      const int m = t * 16 + m15;
      for (int k = 0; k < NLAB; k += 4) {  // 50 K-chunks of 4 labels
        v2f a;
        a.x = diffm[m * NLAB + k + kh];
        a.y = diffm[m * NLAB + k + kh + 1];
        c = __builtin_amdgcn_wmma_f32_16x16x4_f32(
            /*neg_a=*/false, a, /*neg_b=*/false, a,
            /*c_mod=*/(short)0, c, /*reuse_a=*/false, /*reuse_b=*/false);
      }
    }
    // C/D layout: VGPR v -> M=v (lanes 0-15) / M=v+8 (lanes 16-31), N=lane%16.
    // Diagonal (M==N): lanes 0-7 take c[lane], lanes 24-31 take c[lane-24].
    float d = 0.0f;
#pragma unroll
    for (int j = 0; j < 8; ++j) {
      if (L == j || L == j + 24) d = c[j];
    }
    s_diag[L] = d;
  }
  __syncthreads();

  if (tid == 0) {
    float ssq = 0.0f;
    for (int i = 0; i < 32; ++i) ssq += s_diag[i];
    const float mse = ssq / (float)(NHID * NLAB);
    float sum = 0.0f;
    for (int l = 0; l < NLAB; ++l) sum += s_pcn[l];
    float kl = 0.0f;
    for (int l = 0; l < NLAB; ++l) {
      const float p = s_pcn[l] / sum;
      kl += p * (logf(p) - logf(ncells[l]));
    }
    kl /= (float)NLAB;
    out[0] = mse + kl;
  }
}

extern "C" void kernel_launch(void* const* d_in, const int* in_sizes, int n_in,
                              void* d_out, int out_size, void* d_ws, size_t ws_size,
                              hipStream_t stream) {
  const float* pseudo_latent = (const float*)d_in[0];
  const int*   pseudo_labels = (const int*)d_in[1];
  const float* true_latent   = (const float*)d_in[2];
  const int*   true_labels   = (const int*)d_in[3];
  const float* cp            = (const float*)d_in[4];
  const float* pc            = (const float*)d_in[5];
  const float* ct            = (const float*)d_in[6];
  const float* tc            = (const float*)d_in[7];
  const float* ncells        = (const float*)d_in[8];
  float* out = (float*)d_out;
  const int nrows = in_sizes[1];   // 1,000,000 labels

  // Workspace layout: [2][B][SSEG] partials, then [2*SSEG] reduced sums.
  long maxB = ((long)(ws_size / 4) - 2L * SSEG) / (2L * SSEG);
  if (maxB < 1) maxB = 1;
  int B = (int)(maxB < 512 ? maxB : 512);

  float* ws      = (float*)d_ws;
  float* part_p  = ws;
  float* part_t  = ws + (long)B * SSEG;
  float* reduced = ws + 2L * (long)B * SSEG;

  hipLaunchKernelGGL(seg_partial, dim3(B), dim3(256), 0, stream,
                     pseudo_latent, pseudo_labels, part_p, nrows);
  hipLaunchKernelGGL(seg_partial, dim3(B), dim3(256), 0, stream,
                     true_latent, true_labels, part_t, nrows);
  hipLaunchKernelGGL(reduce_partials, dim3((2 * SSEG + 255) / 256), dim3(256), 0, stream,
                     ws, reduced, B);
  hipLaunchKernelGGL(finalize, dim3(1), dim3(256), 0, stream,
                     reduced, cp, pc, ct, tc, ncells, out);
}